// pos_H_MT_version_2_18622978195606
// MI455X (gfx1250) — compile-verified
//
#include <hip/hip_runtime.h>

// Spiking multi-bit quantizer temporal scan (T=4, n_thresh=8).
// Pure HBM-streaming op: 77 MB in + 77 MB out -> ~6.6 us floor @ 23.3 TB/s.
// No matrix op exists in this workload, so no WMMA path; the MI455X win is
// 128-bit coalesced NT loads/stores and maximum memory-level parallelism.

typedef __attribute__((ext_vector_type(4))) float f32x4;

static constexpr int   T_STEPS = 4;     // timesteps (reference T)
static constexpr float SCALE   = 128.f; // 2^(n-1), n = 8 thresholds
static constexpr int   MAX_K   = 255;   // 2^n - 1

__device__ __forceinline__ float quant_step(float& mem, float x, float lam, float rls) {
    float m = mem + x;
    int   k = (int)(m * rls);               // v_cvt_i32_f32: trunc toward zero (== astype(int32))
    k       = k < 0 ? 0 : (k > MAX_K ? MAX_K : k);
    float s = lam * (float)k;               // Σ lam·2^j·bit_j(k) == lam·k
    mem     = m - s;                        // soft reset
    return s;
}

__global__ __launch_bounds__(256) void snn_scan_vec4(const float* __restrict__ x,
                                                     const float* __restrict__ lam_p,
                                                     float* __restrict__ out,
                                                     long long r4,   // #float4 per timestep
                                                     long long R)    // elements per timestep
{
    long long i = (long long)blockIdx.x * blockDim.x + threadIdx.x;
    if (i >= r4) return;

    const float lam = lam_p[0];             // uniform -> scalar load
    const float rls = SCALE / lam;          // one divide per thread (exact for pow2 lam)
    const long long base = i * 4;

    // Independent loads for all 4 timesteps issued back-to-back:
    // 4 outstanding global_load_b128 (TH=NT) per thread for latency hiding.
    const f32x4 x0 = __builtin_nontemporal_load((const f32x4*)(x + 0 * R + base));
    const f32x4 x1 = __builtin_nontemporal_load((const f32x4*)(x + 1 * R + base));
    const f32x4 x2 = __builtin_nontemporal_load((const f32x4*)(x + 2 * R + base));
    const f32x4 x3 = __builtin_nontemporal_load((const f32x4*)(x + 3 * R + base));

    f32x4 mem;
    mem.x = mem.y = mem.z = mem.w = 0.5f * lam;

    f32x4 s0, s1, s2, s3;
#pragma unroll
    for (int c = 0; c < 4; ++c) {
        float m = mem[c];
        s0[c] = quant_step(m, x0[c], lam, rls);
        s1[c] = quant_step(m, x1[c], lam, rls);
        s2[c] = quant_step(m, x2[c], lam, rls);
        s3[c] = quant_step(m, x3[c], lam, rls);
    }

    // Streaming stores (TH=NT): output is never re-read on device.
    __builtin_nontemporal_store(s0, (f32x4*)(out + 0 * R + base));
    __builtin_nontemporal_store(s1, (f32x4*)(out + 1 * R + base));
    __builtin_nontemporal_store(s2, (f32x4*)(out + 2 * R + base));
    __builtin_nontemporal_store(s3, (f32x4*)(out + 3 * R + base));
}

// Scalar tail for R % 4 != 0 (not hit for the given shapes, kept for generality).
__global__ void snn_scan_tail(const float* __restrict__ x,
                              const float* __restrict__ lam_p,
                              float* __restrict__ out,
                              long long start, long long R)
{
    long long r = start + (long long)blockIdx.x * blockDim.x + threadIdx.x;
    if (r >= R) return;
    const float lam = lam_p[0];
    const float rls = SCALE / lam;
    float mem = 0.5f * lam;
#pragma unroll
    for (int t = 0; t < T_STEPS; ++t) {
        float s = quant_step(mem, x[(long long)t * R + r], lam, rls);
        out[(long long)t * R + r] = s;
    }
}

extern "C" void kernel_launch(void* const* d_in, const int* in_sizes, int n_in,
                              void* d_out, int out_size, void* d_ws, size_t ws_size,
                              hipStream_t stream) {
    const float* x     = (const float*)d_in[0];
    const float* lam_p = (const float*)d_in[1];
    float*       out   = (float*)d_out;

    const long long total = (long long)in_sizes[0];   // T*B*TOK*DIM
    const long long R     = total / T_STEPS;          // elements per timestep
    const long long r4    = R / 4;
    const long long rem   = R - r4 * 4;

    if (r4 > 0) {
        const int  block = 256;                       // 8 wave32 waves / block
        const long long grid = (r4 + block - 1) / block;
        snn_scan_vec4<<<(unsigned)grid, block, 0, stream>>>(x, lam_p, out, r4, R);
    }
    if (rem > 0) {
        snn_scan_tail<<<1, 256, 0, stream>>>(x, lam_p, out, r4 * 4, R);
    }
}